// Warp_17927193494409
// MI455X (gfx1250) — compile-verified
//
#include <hip/hip_runtime.h>

// Problem constants
#define BATCH 8
#define CH    32
#define H     256
#define W     256
#define HF    64
#define PAD   25
#define HP    306   // padded size

typedef __attribute__((ext_vector_type(2))) float v2f;
typedef __attribute__((ext_vector_type(8))) float v8f;

// Bicubic kernel pieces, A = -0.75
__device__ __forceinline__ float cc1f(float x) {
    // ((A+2)*x - (A+3))*x*x + 1
    return ((1.25f * x - 2.25f) * x) * x + 1.0f;
}
__device__ __forceinline__ float cc2f(float x) {
    // ((A*x - 5A)*x + 8A)*x - 4A
    return ((-0.75f * x + 3.75f) * x - 6.0f) * x + 3.0f;
}

// ---------------------------------------------------------------------------
// Kernel 0: build the dense 256x64 bicubic resize matrix (align_corners=True,
// border-replicate taps accumulate like the reference's scatter-add).
// One thread per matrix element; 16384 threads.
// ---------------------------------------------------------------------------
__global__ void build_resize_matrix(float* __restrict__ M) {
    int tid = blockIdx.x * blockDim.x + threadIdx.x;   // 0..16383
    int col = tid & 63;
    int row = tid >> 6;
    float src = (float)row * (63.0f / 255.0f);
    float i0f = floorf(src);
    float t   = src - i0f;
    int   i0  = (int)i0f;
    float w0 = cc2f(t + 1.0f);
    float w1 = cc1f(t);
    float w2 = cc1f(1.0f - t);
    float w3 = cc2f(2.0f - t);
    float acc = 0.0f;
    int idx;
    idx = min(max(i0 - 1, 0), 63); if (idx == col) acc += w0;
    idx = min(max(i0    , 0), 63); if (idx == col) acc += w1;
    idx = min(max(i0 + 1, 0), 63); if (idx == col) acc += w2;
    idx = min(max(i0 + 2, 0), 63); if (idx == col) acc += w3;
    M[tid] = acc;
}

// ---------------------------------------------------------------------------
// Kernel 1: T[bc] = M (256x64) @ flow[bc] (64x64)  -> (256x64)
// One wave per 16x16 output tile, WMMA F32 16x16x4, K stepped by 4.
// Waves: 16 bc * (16 * 4) tiles = 1024 waves; 4 waves/block -> 256 blocks.
// ---------------------------------------------------------------------------
__global__ void flow_gemm1(const float* __restrict__ M,
                           const float* __restrict__ flow,
                           float* __restrict__ T) {
    int wave = (int)((blockIdx.x * blockDim.x + threadIdx.x) >> 5);
    int lane = threadIdx.x & 31;
    int bc   = wave >> 6;          // 0..15  (b*2+ch)
    int tile = wave & 63;
    int tm   = tile >> 2;          // 0..15  M-tile
    int tn   = tile & 3;           // 0..3   N-tile
    int r    = lane & 15;
    int kb   = (lane >> 4) << 1;   // lanes 0-15 -> K{0,1}; lanes 16-31 -> K{2,3}

    const float* Arow = M    + (tm * 16 + r) * 64;              // A[row][k]
    const float* Bcol = flow + bc * (64 * 64) + (tn * 16 + r);  // B[k][n], n = r

    v8f c = {};
#pragma unroll
    for (int kk = 0; kk < 16; ++kk) {
        int k0 = kk * 4 + kb;
        v2f a, b;
        a.x = Arow[k0];
        a.y = Arow[k0 + 1];
        b.x = Bcol[k0 * 64];
        b.y = Bcol[(k0 + 1) * 64];
        c = __builtin_amdgcn_wmma_f32_16x16x4_f32(false, a, false, b,
                                                  (short)0, c, false, false);
    }
    // C/D layout: VGPR v -> row v + 8*(lane>=16), col = lane&15
    float* outp = T + bc * (256 * 64) + tn * 16 + (lane & 15);
    int mbase = tm * 16 + ((lane >> 4) << 3);
#pragma unroll
    for (int v = 0; v < 8; ++v)
        outp[(mbase + v) * 64] = c[v];
}

// ---------------------------------------------------------------------------
// Kernel 2: flow_r[bc] = T[bc] (256x64) @ M^T (64x256) -> (256x256)
// B[k][n] = M[n][k] so the per-lane B fetch is a contiguous pair from M.
// Waves: 16 bc * 256 tiles = 4096 waves; 4 waves/block -> 1024 blocks.
// ---------------------------------------------------------------------------
__global__ void flow_gemm2(const float* __restrict__ T,
                           const float* __restrict__ M,
                           float* __restrict__ FR) {
    int wave = (int)((blockIdx.x * blockDim.x + threadIdx.x) >> 5);
    int lane = threadIdx.x & 31;
    int bc   = wave >> 8;          // 0..15
    int tile = wave & 255;
    int tm   = tile >> 4;          // 0..15
    int tn   = tile & 15;          // 0..15
    int r    = lane & 15;
    int kb   = (lane >> 4) << 1;

    const float* Arow = T + bc * (256 * 64) + (tm * 16 + r) * 64;
    const float* Brow = M + (tn * 16 + r) * 64;   // M[n][k] contiguous in k

    v8f c = {};
#pragma unroll
    for (int kk = 0; kk < 16; ++kk) {
        int k0 = kk * 4 + kb;
        v2f a, b;
        a.x = Arow[k0];
        a.y = Arow[k0 + 1];
        b.x = Brow[k0];
        b.y = Brow[k0 + 1];
        c = __builtin_amdgcn_wmma_f32_16x16x4_f32(false, a, false, b,
                                                  (short)0, c, false, false);
    }
    float* outp = FR + bc * (256 * 256) + (tm * 16) * 256 + tn * 16 + (lane & 15);
    int mofs = (lane >> 4) << 3;
#pragma unroll
    for (int v = 0; v < 8; ++v)
        outp[(mofs + v) * 256] = c[v];
}

// ---------------------------------------------------------------------------
// Kernel 3: bicubic grid-sample warp. One thread per (b,i,j); the 16 weights
// and 16 gather offsets are channel-invariant, so compute once and sweep all
// 32 channels. Zero-weight taps are still loaded from a clamped (valid)
// address to stay branch-free.
// ---------------------------------------------------------------------------
__global__ void warp_kernel(const float* __restrict__ input,
                            const float* __restrict__ FR,
                            float* __restrict__ out) {
    int tid = blockIdx.x * blockDim.x + threadIdx.x;  // 0..524287
    int j = tid & 255;
    int i = (tid >> 8) & 255;
    int b = tid >> 16;

    float fx = FR[((b * 2 + 0) * 256 + i) * 256 + j];
    float fy = FR[((b * 2 + 1) * 256 + i) * 256 + j];

    // g = flow_r - grid256 + grid306(shifted by PAD); then to pixel coords
    float g0 = fx - (-1.0f + (float)j * (2.0f / 255.0f))
                  + (-1.0f + (float)(j + PAD) * (2.0f / 305.0f));
    float g1 = fy - (-1.0f + (float)i * (2.0f / 255.0f))
                  + (-1.0f + (float)(i + PAD) * (2.0f / 305.0f));
    float x = (g0 + 1.0f) * 0.5f * 305.0f;
    float y = (g1 + 1.0f) * 0.5f * 305.0f;

    float xf = floorf(x), yf = floorf(y);
    float tx = x - xf,    ty = y - yf;
    int   ix = (int)xf,   iy = (int)yf;

    float wxv[4] = { cc2f(tx + 1.0f), cc1f(tx), cc1f(1.0f - tx), cc2f(2.0f - tx) };
    float wyv[4] = { cc2f(ty + 1.0f), cc1f(ty), cc1f(1.0f - ty), cc2f(2.0f - ty) };

    int   xo[4], yo[4];
    float wxm[4], wym[4];
#pragma unroll
    for (int n = 0; n < 4; ++n) {
        int xx = ix - 1 + n;                       // padded coords
        wxm[n] = (xx >= 0 && xx < HP) ? wxv[n] : 0.0f;   // zeros padding
        xo[n]  = min(max(xx - PAD, 0), W - 1);           // edge replicate
        int yy = iy - 1 + n;
        wym[n] = (yy >= 0 && yy < HP) ? wyv[n] : 0.0f;
        yo[n]  = min(max(yy - PAD, 0), H - 1) * W;
    }

    float w16[16];
    int   o16[16];
#pragma unroll
    for (int m = 0; m < 4; ++m)
#pragma unroll
        for (int n = 0; n < 4; ++n) {
            w16[m * 4 + n] = wym[m] * wxm[n];
            o16[m * 4 + n] = yo[m] + xo[n];
        }

    const float* base  = input + (size_t)b * (CH * H * W);
    float*       obase = out   + (size_t)b * (CH * H * W) + i * W + j;
#pragma unroll 4
    for (int c = 0; c < CH; ++c) {
        const float* p = base + c * (H * W);
        float acc = 0.0f;
#pragma unroll
        for (int k = 0; k < 16; ++k)
            acc = fmaf(w16[k], p[o16[k]], acc);
        obase[c * (H * W)] = acc;
    }
}

// ---------------------------------------------------------------------------
extern "C" void kernel_launch(void* const* d_in, const int* in_sizes, int n_in,
                              void* d_out, int out_size, void* d_ws, size_t ws_size,
                              hipStream_t stream) {
    const float* input = (const float*)d_in[0];   // [8,32,256,256]
    const float* flow  = (const float*)d_in[1];   // [8,2,64,64]
    float*       out   = (float*)d_out;           // [8,32,256,256]

    float* ws = (float*)d_ws;
    float* M  = ws;                        // 256*64            = 16384 floats
    float* T  = ws + 16384;                // 16*256*64         = 262144 floats
    float* FR = ws + 16384 + 262144;       // 16*256*256        = 1048576 floats
    // total workspace: ~5.1 MB

    build_resize_matrix<<<64, 256, 0, stream>>>(M);
    flow_gemm1<<<256, 128, 0, stream>>>(M, flow, T);
    flow_gemm2<<<1024, 128, 0, stream>>>(T, M, FR);
    warp_kernel<<<2048, 256, 0, stream>>>(input, FR, out);
}